// DTMEmbedding_22436909154637
// MI455X (gfx1250) — compile-verified
//
#include <hip/hip_runtime.h>

#define N_PTS 2048
#define BATCH 8
#define KNN 32
#define BN_EPS 1e-5f
#define SLOPE 0.2f

typedef __attribute__((ext_vector_type(2))) float v2f;
typedef __attribute__((ext_vector_type(8))) float v8f;

__device__ __forceinline__ float lrelu(float y) { return (y >= 0.f) ? y : SLOPE * y; }

// Pair-interleaved LDS layout for WMMA B operands:
// element (k, col) lives at pair index (k>>1)*32 + col, component (k&1).
// A B-operand K-pair is then a single aligned 8-byte LDS load.
__device__ __forceinline__ int pidx(int k, int col) {
  return ((((k >> 1) * 32) + col) << 1) | (k & 1);
}

// ---------------------------------------------------------------------------
// Kernel 1: DTM. One wave per row i: distances to all j, radix-select the
// 1024th smallest + sum of the 1024 smallest, then dtm = sqrt(cum'/wb).
// ---------------------------------------------------------------------------
__global__ __launch_bounds__(128) void dtm_kernel(const float* __restrict__ x,
                                                  const float* __restrict__ m0p,
                                                  float* __restrict__ h) {
  __shared__ float xb[3][N_PTS];   // 24 KB
  __shared__ float drow[4][N_PTS]; // 32 KB
  const int b = blockIdx.x >> 9;            // 512 blocks per batch
  const int row0 = (blockIdx.x & 511) * 4;
  const int t = threadIdx.x, wave = t >> 5, lane = t & 31;

  for (int e = t; e < 3 * N_PTS; e += 128)
    xb[e >> 11][e & (N_PTS - 1)] = x[(size_t)b * 3 * N_PTS + e];
  __syncthreads();

  const int i = row0 + wave;
  const float xi0 = xb[0][i], xi1 = xb[1][i], xi2 = xb[2][i];
  for (int j = lane; j < N_PTS; j += 32) {
    float d0 = xi0 - xb[0][j], d1 = xi1 - xb[1][j], d2 = xi2 - xb[2][j];
    drow[wave][j] = sqrtf(d0 * d0 + d1 * d1 + d2 * d2);
  }
  __syncthreads();

  const int kth = 1024;  // ceil(N*M0_INIT)
  unsigned p = 0u;       // radix-select kth smallest (uint order == float order, x>=0)
  for (int bit = 30; bit >= 0; --bit) {
    unsigned c = p | (1u << bit);
    int cnt = 0;
    for (int j = lane; j < N_PTS; j += 32)
      cnt += (__float_as_uint(drow[wave][j]) < c) ? 1 : 0;
    for (int off = 16; off > 0; off >>= 1) cnt += __shfl_xor(cnt, off, 32);
    if (cnt < kth) p = c;
  }
  const float tval = __uint_as_float(p);
  int cntLess = 0; float sumLess = 0.f;
  for (int j = lane; j < N_PTS; j += 32) {
    float dv = drow[wave][j];
    if (__float_as_uint(dv) < p) { cntLess++; sumLess += dv; }
  }
  for (int off = 16; off > 0; off >>= 1) {
    cntLess += __shfl_xor(cntLess, off, 32);
    sumLess += __shfl_xor(sumLess, off, 32);
  }
  if (lane == 0) {
    float m0 = m0p[0];
    float wb = (float)N_PTS * m0;
    float wsum = ceilf(wb);
    float cum = sumLess + (float)(kth - cntLess) * tval;   // sum of 1024 smallest
    float dtm = cum + tval * (wb - wsum);
    h[(size_t)b * N_PTS + i] = sqrtf(dtm / wb);            // R = 2
  }
}

// ---------------------------------------------------------------------------
// Kernel 2: 32-NN on 1-D dtm values. One wave per row. Radix-select the 32nd
// smallest d2, then ballot-compact indices (strictly-less first, then ties).
// ---------------------------------------------------------------------------
__global__ __launch_bounds__(256) void knn1_kernel(const float* __restrict__ h,
                                                   unsigned short* __restrict__ idx1) {
  __shared__ float hb[N_PTS];
  const int b = blockIdx.x >> 8;            // 256 blocks per batch
  const int row0 = (blockIdx.x & 255) * 8;
  const int t = threadIdx.x, wave = t >> 5, lane = t & 31;
  for (int e = t; e < N_PTS; e += 256) hb[e] = h[(size_t)b * N_PTS + e];
  __syncthreads();

  const int i = row0 + wave;
  const float hi = hb[i];
  unsigned p = 0u;
  for (int bit = 30; bit >= 0; --bit) {
    unsigned c = p | (1u << bit);
    int cnt = 0;
    for (int j = lane; j < N_PTS; j += 32) {
      float df = hi - hb[j];
      cnt += (__float_as_uint(df * df) < c) ? 1 : 0;
    }
    for (int off = 16; off > 0; off >>= 1) cnt += __shfl_xor(cnt, off, 32);
    if (cnt < KNN) p = c;
  }
  unsigned short* out = idx1 + ((size_t)b * N_PTS + i) * KNN;
  int written = 0;
  for (int base = 0; base < N_PTS; base += 32) {       // strictly less: count <= 31
    int j = base + lane;
    float df = hi - hb[j];
    bool pred = (__float_as_uint(df * df) < p);
    unsigned mm = (unsigned)__ballot(pred);
    int pos = written + __popc(mm & ((1u << lane) - 1u));
    if (pred) out[pos] = (unsigned short)j;
    written += __popc(mm);
  }
  for (int base = 0; base < N_PTS && written < KNN; base += 32) {  // fill ties
    int j = base + lane;
    float df = hi - hb[j];
    bool pred = (__float_as_uint(df * df) == p);
    unsigned mm = (unsigned)__ballot(pred);
    int pos = written + __popc(mm & ((1u << lane) - 1u));
    if (pred && pos < KNN) out[pos] = (unsigned short)j;
    written += __popc(mm);
  }
}

// ---------------------------------------------------------------------------
// Kernel 3: stage-1 edgeconv, fused per point. Block = 4 waves = 1 point.
// conv1 (K=2) in VALU -> H1 (128x32, pair-interleaved) in LDS; conv2
// (64x128 @ 128x32) via V_WMMA_F32_16X16X4_F32; BN+lrelu; max over k ->
// f1 into d_out ch 0..63.
// ---------------------------------------------------------------------------
__global__ __launch_bounds__(128) void stage1_kernel(
    const float* __restrict__ h, const unsigned short* __restrict__ idx1,
    const float* __restrict__ w1, const float* __restrict__ g1, const float* __restrict__ bb1,
    const float* __restrict__ rm1, const float* __restrict__ rv1,
    const float* __restrict__ w2, const float* __restrict__ g2, const float* __restrict__ bb2,
    const float* __restrict__ rm2, const float* __restrict__ rv2,
    float* __restrict__ out) {
  __shared__ v2f H1[64 * 32];     // 128x32 floats, pair-interleaved (16 KB)
  __shared__ float nbv[KNN];
  __shared__ float sc2[64], sh2[64];
  const int b = blockIdx.x >> 11, n = blockIdx.x & (N_PTS - 1);
  const int t = threadIdx.x, wave = t >> 5, lane = t & 31, lm = lane & 15;
  const float cen = h[(size_t)b * N_PTS + n];

  if (t < KNN) nbv[t] = h[(size_t)b * N_PTS + idx1[((size_t)b * N_PTS + n) * KNN + t]];
  if (t < 64) {
    float s = g2[t] * rsqrtf(rv2[t] + BN_EPS);
    sc2[t] = s; sh2[t] = bb2[t] - rm2[t] * s;
  }
  __syncthreads();

  // conv1 + BN1 + lrelu -> H1[(o,k)] pair-interleaved
  for (int e = t; e < 128 * KNN; e += 128) {
    int o = e >> 5, k = e & 31;
    float s1 = g1[o] * rsqrtf(rv1[o] + BN_EPS);
    float y = w1[o * 2 + 0] * cen + w1[o * 2 + 1] * (nbv[k] - cen);
    y = y * s1 + (bb1[o] - rm1[o] * s1);
    ((float*)H1)[pidx(o, k)] = lrelu(y);
  }
  __syncthreads();

  // conv2: wave = mTile (rows 16w..16w+15), two nTiles of 16 k-columns
  const int halfK = (lane < 16) ? 0 : 2;
  const int mrow = wave * 16 + lm;
  v8f acc0 = {}, acc1 = {};
  for (int kk = 0; kk < 32; ++kk) {
    int k0 = kk * 4 + halfK;
    int kp = (k0 >> 1) * 32;                    // pair-row base
    v2f a;  a.x = w2[mrow * 128 + k0]; a.y = w2[mrow * 128 + k0 + 1];
    v2f b0 = H1[kp + lm];                       // (k0, col) & (k0+1, col)
    v2f b1 = H1[kp + 16 + lm];
    acc0 = __builtin_amdgcn_wmma_f32_16x16x4_f32(false, a, false, b0, (short)0, acc0, false, false);
    acc1 = __builtin_amdgcn_wmma_f32_16x16x4_f32(false, a, false, b1, (short)0, acc1, false, false);
  }
  for (int v = 0; v < 8; ++v) {
    int ch = wave * 16 + ((lane < 16) ? v : v + 8);
    float y0 = lrelu(acc0[v] * sc2[ch] + sh2[ch]);
    float y1 = lrelu(acc1[v] * sc2[ch] + sh2[ch]);
    float r = fmaxf(y0, y1);
    for (int off = 1; off < 16; off <<= 1) r = fmaxf(r, __shfl_xor(r, off, 32));
    if (lm == 0) out[((size_t)b * 128 + ch) * N_PTS + n] = r;  // f1 -> ch 0..63
  }
}

// ---------------------------------------------------------------------------
// Kernel 4: 32-NN on 64-D f1 (read from d_out ch 0..63). One wave per row.
// ---------------------------------------------------------------------------
__global__ __launch_bounds__(128) void knn2_kernel(const float* __restrict__ outIn,
                                                   unsigned short* __restrict__ idx2) {
  __shared__ float fc[4][64];
  __shared__ float d2row[4][N_PTS];  // 32 KB
  const int b = blockIdx.x >> 9, row0 = (blockIdx.x & 511) * 4;
  const int t = threadIdx.x, wave = t >> 5, lane = t & 31;
  const int i = row0 + wave;
  const float* f1b = outIn + (size_t)b * 128 * N_PTS;

  fc[wave][lane]      = f1b[(size_t)lane * N_PTS + i];
  fc[wave][lane + 32] = f1b[(size_t)(lane + 32) * N_PTS + i];
  __syncthreads();

  for (int j = lane; j < N_PTS; j += 32) {
    float acc = 0.f;
    for (int c = 0; c < 64; ++c) {
      float df = fc[wave][c] - f1b[(size_t)c * N_PTS + j];
      acc += df * df;
    }
    d2row[wave][j] = acc;
  }
  __syncthreads();

  unsigned p = 0u;
  for (int bit = 30; bit >= 0; --bit) {
    unsigned c = p | (1u << bit);
    int cnt = 0;
    for (int j = lane; j < N_PTS; j += 32)
      cnt += (__float_as_uint(d2row[wave][j]) < c) ? 1 : 0;
    for (int off = 16; off > 0; off >>= 1) cnt += __shfl_xor(cnt, off, 32);
    if (cnt < KNN) p = c;
  }
  unsigned short* out = idx2 + ((size_t)b * N_PTS + i) * KNN;
  int written = 0;
  for (int base = 0; base < N_PTS; base += 32) {
    int j = base + lane;
    bool pred = (__float_as_uint(d2row[wave][j]) < p);
    unsigned mm = (unsigned)__ballot(pred);
    int pos = written + __popc(mm & ((1u << lane) - 1u));
    if (pred) out[pos] = (unsigned short)j;
    written += __popc(mm);
  }
  for (int base = 0; base < N_PTS && written < KNN; base += 32) {
    int j = base + lane;
    bool pred = (__float_as_uint(d2row[wave][j]) == p);
    unsigned mm = (unsigned)__ballot(pred);
    int pos = written + __popc(mm & ((1u << lane) - 1u));
    if (pred && pos < KNN) out[pos] = (unsigned short)j;
    written += __popc(mm);
  }
}

// ---------------------------------------------------------------------------
// Kernel 5: stage-2 edgeconv, fused per point. H2 (128x32) in LDS; conv3
// (128x128) WMMA -> BN+lrelu -> Y3 in LDS; conv4 (64x128) WMMA -> BN+lrelu
// -> max over k -> f2 into d_out ch 64..127. H2/Y3 pair-interleaved.
// ---------------------------------------------------------------------------
__global__ __launch_bounds__(128) void stage2_kernel(
    const float* __restrict__ outIn, const unsigned short* __restrict__ idx2,
    const float* __restrict__ w3, const float* __restrict__ g3, const float* __restrict__ bb3,
    const float* __restrict__ rm3, const float* __restrict__ rv3,
    const float* __restrict__ w4, const float* __restrict__ g4, const float* __restrict__ bb4,
    const float* __restrict__ rm4, const float* __restrict__ rv4,
    float* __restrict__ out) {
  __shared__ v2f H2[64 * 32];     // 128x32 floats, pair-interleaved (16 KB)
  __shared__ v2f Y3[64 * 32];     // 128x32 floats, pair-interleaved (16 KB)
  __shared__ float sc4[64], sh4[64];
  const int b = blockIdx.x >> 11, n = blockIdx.x & (N_PTS - 1);
  const int t = threadIdx.x, wave = t >> 5, lane = t & 31, lm = lane & 15;
  const float* f1b = outIn + (size_t)b * 128 * N_PTS;
  const unsigned short* nb = idx2 + ((size_t)b * N_PTS + n) * KNN;

  for (int e = t; e < 64 * KNN; e += 128) {
    int c = e >> 5, k = e & 31;
    float cen = f1b[(size_t)c * N_PTS + n];
    float nv = f1b[(size_t)c * N_PTS + nb[k]];
    ((float*)H2)[pidx(c, k)] = cen;             // channels 0..63: center
    ((float*)H2)[pidx(c + 64, k)] = nv - cen;   // channels 64..127: diff
  }
  if (t < 64) {
    float s = g4[t] * rsqrtf(rv4[t] + BN_EPS);
    sc4[t] = s; sh4[t] = bb4[t] - rm4[t] * s;
  }
  __syncthreads();

  const int halfK = (lane < 16) ? 0 : 2;
  // conv3: 8 mTiles x 2 nTiles; wave handles mTiles {wave, wave+4}
  for (int mt = wave; mt < 8; mt += 4) {
    v8f a0 = {}, a1 = {};
    const int mrow = mt * 16 + lm;
    for (int kk = 0; kk < 32; ++kk) {
      int k0 = kk * 4 + halfK;
      int kp = (k0 >> 1) * 32;
      v2f a;  a.x = w3[mrow * 128 + k0]; a.y = w3[mrow * 128 + k0 + 1];
      v2f b0 = H2[kp + lm];
      v2f b1 = H2[kp + 16 + lm];
      a0 = __builtin_amdgcn_wmma_f32_16x16x4_f32(false, a, false, b0, (short)0, a0, false, false);
      a1 = __builtin_amdgcn_wmma_f32_16x16x4_f32(false, a, false, b1, (short)0, a1, false, false);
    }
    for (int v = 0; v < 8; ++v) {
      int ch = mt * 16 + ((lane < 16) ? v : v + 8);
      float s = g3[ch] * rsqrtf(rv3[ch] + BN_EPS);
      float sh = bb3[ch] - rm3[ch] * s;
      ((float*)Y3)[pidx(ch, lm)]      = lrelu(a0[v] * s + sh);
      ((float*)Y3)[pidx(ch, 16 + lm)] = lrelu(a1[v] * s + sh);
    }
  }
  __syncthreads();

  // conv4: wave = mTile 0..3
  v8f c0 = {}, c1 = {};
  const int mrow4 = wave * 16 + lm;
  for (int kk = 0; kk < 32; ++kk) {
    int k0 = kk * 4 + halfK;
    int kp = (k0 >> 1) * 32;
    v2f a;  a.x = w4[mrow4 * 128 + k0]; a.y = w4[mrow4 * 128 + k0 + 1];
    v2f b0 = Y3[kp + lm];
    v2f b1 = Y3[kp + 16 + lm];
    c0 = __builtin_amdgcn_wmma_f32_16x16x4_f32(false, a, false, b0, (short)0, c0, false, false);
    c1 = __builtin_amdgcn_wmma_f32_16x16x4_f32(false, a, false, b1, (short)0, c1, false, false);
  }
  for (int v = 0; v < 8; ++v) {
    int ch = wave * 16 + ((lane < 16) ? v : v + 8);
    float y0 = lrelu(c0[v] * sc4[ch] + sh4[ch]);
    float y1 = lrelu(c1[v] * sc4[ch] + sh4[ch]);
    float r = fmaxf(y0, y1);
    for (int off = 1; off < 16; off <<= 1) r = fmaxf(r, __shfl_xor(r, off, 32));
    if (lm == 0) out[((size_t)b * 128 + 64 + ch) * N_PTS + n] = r;  // f2 -> ch 64..127
  }
}

// ---------------------------------------------------------------------------
extern "C" void kernel_launch(void* const* d_in, const int* in_sizes, int n_in,
                              void* d_out, int out_size, void* d_ws, size_t ws_size,
                              hipStream_t stream) {
  const float* x   = (const float*)d_in[0];
  const float* m0  = (const float*)d_in[1];
  const float* w1  = (const float*)d_in[2];
  const float* g1  = (const float*)d_in[3];
  const float* bb1 = (const float*)d_in[4];
  const float* rm1 = (const float*)d_in[5];
  const float* rv1 = (const float*)d_in[6];
  const float* w2  = (const float*)d_in[7];
  const float* g2  = (const float*)d_in[8];
  const float* bb2 = (const float*)d_in[9];
  const float* rm2 = (const float*)d_in[10];
  const float* rv2 = (const float*)d_in[11];
  const float* w3  = (const float*)d_in[12];
  const float* g3  = (const float*)d_in[13];
  const float* bb3 = (const float*)d_in[14];
  const float* rm3 = (const float*)d_in[15];
  const float* rv3 = (const float*)d_in[16];
  const float* w4  = (const float*)d_in[17];
  const float* g4  = (const float*)d_in[18];
  const float* bb4 = (const float*)d_in[19];
  const float* rm4 = (const float*)d_in[20];
  const float* rv4 = (const float*)d_in[21];
  float* out = (float*)d_out;

  char* ws = (char*)d_ws;
  float* h = (float*)ws;                                             // 64 KB
  unsigned short* idx1 = (unsigned short*)(ws + 65536);              // 1 MB
  unsigned short* idx2 = (unsigned short*)(ws + 65536 + (size_t)BATCH * N_PTS * KNN * 2);

  dtm_kernel   <<<BATCH * (N_PTS / 4), 128, 0, stream>>>(x, m0, h);
  knn1_kernel  <<<BATCH * (N_PTS / 8), 256, 0, stream>>>(h, idx1);
  stage1_kernel<<<BATCH * N_PTS, 128, 0, stream>>>(h, idx1, w1, g1, bb1, rm1, rv1,
                                                   w2, g2, bb2, rm2, rv2, out);
  knn2_kernel  <<<BATCH * (N_PTS / 4), 128, 0, stream>>>(out, idx2);
  stage2_kernel<<<BATCH * N_PTS, 128, 0, stream>>>(out, idx2, w3, g3, bb3, rm3, rv3,
                                                   w4, g4, bb4, rm4, rv4, out);
}